// NUFFTLayerMultiChannel3D_65910568125080
// MI455X (gfx1250) — compile-verified
//
#include <hip/hip_runtime.h>
#include <math.h>

// ---------------- CDNA5 WMMA / TDM types ----------------
typedef __attribute__((ext_vector_type(2))) float v2f;
typedef __attribute__((ext_vector_type(8))) float v8f;
typedef __attribute__((ext_vector_type(4))) unsigned int u32x4;
typedef __attribute__((ext_vector_type(4))) int i32x4;
typedef __attribute__((ext_vector_type(8))) int i32x8;

// ---------------- problem constants ----------------
#define NM    65           // mesh points per axis
#define NM2   4225         // 65*65
#define NMP   66           // padded axis length (one zero plane)
#define VOLP  287496       // 66^3 floats per complex-plane volume
#define IJP   4240         // 265*16, padded ij row space
#define BATCH 8
#define PPTS  256
#define TWLD  80           // padded twiddle leading dim (5*16)
#define NTILES 1325        // 265 row-tiles * 5 col-tiles

#define PI_F  3.14159265358979f
#define L_F   6.28318530717959f
#define TAU_F (12.0f / 4225.0f)        // 12*(L/(2*pi*N))^2 with L=2*pi
#define INV4T (4225.0f / 48.0f)        // 1/(4*tau)

// ---------------- workspace layout (float offsets) ----------------
#define OFF_DFRE 0
#define OFF_DFIM 6400
#define OFF_DIRE 12800
#define OFF_DIIM 19200
#define OFF_G     25600                          // [B][3][P][80]
#define OFF_GXY   (OFF_G + BATCH*3*PPTS*80)      // [B][P][IJP]
#define OFF_BARE  (OFF_GXY + BATCH*PPTS*IJP)     // 16 volumes Re
#define OFF_BAIM  (OFF_BARE + 16*VOLP)
#define OFF_BBRE  (OFF_BAIM + 16*VOLP)
#define OFF_BBIM  (OFF_BBRE + 16*VOLP)
#define OFF_S     (OFF_BBIM + 16*VOLP)           // [B*2][P][80]

// =====================================================================
// TDM: 1-D copy of n contiguous dwords global -> LDS via Tensor Data Mover.
// D# built per CDNA5 ISA ch.8 (count=1, type=2, data_size=4B, 1-D tile).
// =====================================================================
__device__ __forceinline__ void tdm_copy_dwords(const float* __restrict__ src,
                                                void* ldsDst, int ndw) {
  u32x4 g0;
  unsigned long long ga = (unsigned long long)(uintptr_t)src;
  g0[0] = 1u;                                   // count=1, user descriptor
  g0[1] = (unsigned int)(uintptr_t)ldsDst;      // lds_addr (byte offset)
  g0[2] = (unsigned int)(ga & 0xFFFFFFFFu);     // global_addr lo
  g0[3] = (unsigned int)((ga >> 32) & 0x01FFFFFFu) | 0x80000000u; // hi | type=2
  i32x8 g1;
  g1[0] = 0x00020000;                           // data_size=2 (4 bytes), no multicast
  g1[1] = (ndw & 0xFFFF) << 16;                 // tensor_dim0[15:0] in bits[63:48]
  g1[2] = ((ndw >> 16) & 0xFFFF) | (1 << 16);   // tensor_dim0[31:16], tensor_dim1=1
  g1[3] = (ndw & 0xFFFF) << 16;                 // tile_dim0 = ndw
  g1[4] = 1;                                    // tile_dim1=1, tile_dim2=0
  g1[5] = ndw;                                  // tensor_dim0_stride lo
  g1[6] = 0;
  g1[7] = 0;
  i32x4 z4 = {0, 0, 0, 0};
#if defined(__clang_major__) && (__clang_major__ >= 23)
  i32x8 z8 = {0, 0, 0, 0, 0, 0, 0, 0};
  __builtin_amdgcn_tensor_load_to_lds(g0, g1, z4, z4, z8, 0);
#else
  __builtin_amdgcn_tensor_load_to_lds(g0, g1, z4, z4, 0);
#endif
}

// =====================================================================
// Async global->LDS staging of one b128 (4 dwords) per call.
// =====================================================================
__device__ __forceinline__ void async_b128_to_lds(const float* __restrict__ gptr,
                                                  void* ldsDst) {
  unsigned int l = (unsigned int)(uintptr_t)ldsDst;
  asm volatile("global_load_async_to_lds_b128 %0, %1, off"
               :: "v"(l), "v"(gptr) : "memory");
}
__device__ __forceinline__ void wait_async0() {
  asm volatile("s_wait_asynccnt 0x0" ::: "memory");
}

// =====================================================================
// Twiddle init (fftshift folded into forward rows, ifftshift+1/N into inverse)
// =====================================================================
__global__ void init_twiddles(float* __restrict__ ws) {
  int t = blockIdx.x * blockDim.x + threadIdx.x;
  if (t >= TWLD * TWLD) return;
  int c = t / TWLD, n = t % TWLD;
  float dfre = 0.f, dfim = 0.f, dire = 0.f, diim = 0.f;
  if (c < NM && n < NM) {
    int hc = (c + 33) % NM;
    int m1 = (hc * n) % NM;
    float a1 = (2.0f * PI_F / (float)NM) * (float)m1;
    dfre = cosf(a1);
    dfim = -sinf(a1);
    int hn = (n + 33) % NM;
    int m2 = (hn * c) % NM;
    float a2 = (2.0f * PI_F / (float)NM) * (float)m2;
    dire = cosf(a2) * (1.0f / (float)NM);
    diim = sinf(a2) * (1.0f / (float)NM);
  }
  ws[OFF_DFRE + t] = dfre;
  ws[OFF_DFIM + t] = dfim;
  ws[OFF_DIRE + t] = dire;
  ws[OFF_DIIM + t] = diim;
}

__global__ void zero_kernel(float* __restrict__ p, size_t n) {
  size_t i = (size_t)blockIdx.x * blockDim.x + threadIdx.x;
  size_t stride = (size_t)gridDim.x * blockDim.x;
  for (; i < n; i += stride) p[i] = 0.f;
}

// =====================================================================
// Periodic Gaussian factors gx,gy,gz : [B][3][P][80], zero padded cols.
// =====================================================================
__device__ __forceinline__ float gauss_per(float d) {
  float a = d, b = d - L_F, c = d + L_F;
  return expf(-a * a * INV4T) + expf(-b * b * INV4T) + expf(-c * c * INV4T);
}

__global__ void compute_g(const float* __restrict__ pts, float* __restrict__ G) {
  int t = blockIdx.x * blockDim.x + threadIdx.x;   // b*P + p
  if (t >= BATCH * PPTS) return;
  int b = t / PPTS, p = t % PPTS;
  for (int ax = 0; ax < 3; ++ax) {
    float xv = pts[t * 3 + ax];
    float* g = G + ((size_t)((b * 3 + ax) * PPTS + p)) * 80;
    for (int i = 0; i < 80; ++i) {
      if (i < NM) {
        float d = xv - (L_F / (float)NM) * (float)i;
        g[i] = gauss_per(d);
      } else {
        g[i] = 0.f;
      }
    }
  }
}

__global__ void compute_gxy(const float* __restrict__ G, float* __restrict__ GXY) {
  int bp = blockIdx.x;
  int b = bp / PPTS, p = bp % PPTS;
  const float* gx = G + ((size_t)((b * 3 + 0) * PPTS + p)) * 80;
  const float* gy = G + ((size_t)((b * 3 + 1) * PPTS + p)) * 80;
  float* out = GXY + (size_t)bp * IJP;
  for (int ij = threadIdx.x; ij < IJP; ij += blockDim.x) {
    if (ij < NM2) {
      int i = ij / NM, j = ij - i * NM;
      out[ij] = gx[i] * gy[j];
    } else {
      out[ij] = 0.f;
    }
  }
}

// =====================================================================
// Spread (WMMA f32): rho[b][ij][k] = sum_p gxy[b][p][ij] * gz[b][p][k]
// 256-thread block = 8 waves, each wave one 16x16 tile.
// gz panel (256x80) staged in LDS via async global->LDS b128 loads.
// =====================================================================
__global__ void __launch_bounds__(256) spread_wmma(const float* __restrict__ GXY,
                                                   const float* __restrict__ G,
                                                   float* __restrict__ outRe) {
  __shared__ float sGZ[PPTS * 80];
  const int b = blockIdx.z;
  const float* gz = G + ((size_t)(b * 3 + 2) * PPTS) * 80;
  for (int i = threadIdx.x * 4; i < PPTS * 80; i += 256 * 4)
    async_b128_to_lds(gz + i, &sGZ[i]);
  wait_async0();
  __syncthreads();

  const int w = threadIdx.x >> 5;
  const int t = blockIdx.x * 8 + w;
  if (t < NTILES) {
    const int mtile = t / 5, ctile = t - mtile * 5;
    const int lm = threadIdx.x & 15;
    const int ko = ((threadIdx.x >> 4) & 1) << 1;
    const float* A = GXY + (size_t)b * PPTS * IJP;   // [p][ij]
    const int am = mtile * 16 + lm;
    const int bcol = ctile * 16 + lm;

    v8f acc = {0.f, 0.f, 0.f, 0.f, 0.f, 0.f, 0.f, 0.f};
#pragma unroll 4
    for (int k0 = 0; k0 < PPTS; k0 += 4) {
      const int n0 = k0 + ko, n1 = n0 + 1;
      __builtin_prefetch(&A[(size_t)(n0 + 32) * IJP + am], 0, 0);
      v2f a, bb;
      a[0] = A[(size_t)n0 * IJP + am];
      a[1] = A[(size_t)n1 * IJP + am];
      bb[0] = sGZ[n0 * 80 + bcol];
      bb[1] = sGZ[n1 * 80 + bcol];
      acc = __builtin_amdgcn_wmma_f32_16x16x4_f32(false, a, false, bb, (short)0, acc,
                                                  false, false);
    }
    const int cOut = bcol;
    const int rhi = ((threadIdx.x >> 4) & 1) << 3;
    const int rBase = mtile * 16 + rhi;
    int u0 = rBase / NM;
    int v0 = rBase - u0 * NM;
#pragma unroll
    for (int q = 0; q < 8; ++q) {
      int vq = v0 + q;
      int uq = u0 + (vq >= NM ? 1 : 0);
      vq -= (vq >= NM ? NM : 0);
      if ((rBase + q) < NM2 && cOut < NM) {
        outRe[(size_t)b * VOLP + (size_t)uq * (NMP * NMP) + (size_t)vq * NMP + cOut] =
            acc[q];
      }
    }
  }
}

// =====================================================================
// Complex DFT axis pass (WMMA f32): out[r][c] = sum_n in[r][n] * D[c][n]
// 256-thread block = 8 waves; twiddle matrices staged in LDS by the
// Tensor Data Mover (wave 0 issues two 1-D D# descriptors, then
// s_wait_tensorcnt + workgroup barrier). 68 static v_wmma per kernel.
// =====================================================================
__global__ void __launch_bounds__(256) dft_pass_wmma(
    const float* __restrict__ inRe, const float* __restrict__ inIm,
    float* __restrict__ outRe, float* __restrict__ outIm,
    const float* __restrict__ Dre, const float* __restrict__ Dim,
    int sA, int sU, int sV) {
  __shared__ float sDre[TWLD * TWLD];
  __shared__ float sDim[TWLD * TWLD];
  if (threadIdx.x < 32) {
    tdm_copy_dwords(Dre, sDre, TWLD * TWLD);
    tdm_copy_dwords(Dim, sDim, TWLD * TWLD);
    __builtin_amdgcn_s_wait_tensorcnt(0);
  }
  __syncthreads();

  const int w = threadIdx.x >> 5;
  const int t = blockIdx.x * 8 + w;
  if (t < NTILES) {
    const int mtile = t / 5, ctile = t - mtile * 5;
    const int vol = blockIdx.z;
    const size_t vb = (size_t)vol * VOLP;
    const int lm = threadIdx.x & 15;
    const int ko = ((threadIdx.x >> 4) & 1) << 1;

    const int r = mtile * 16 + lm;
    const int u = r / NM, v = r - u * NM;
    const size_t abase = vb + (size_t)u * sU + (size_t)v * sV;
    const int drow = (ctile * 16 + lm) * TWLD;

    v8f accR = {0.f, 0.f, 0.f, 0.f, 0.f, 0.f, 0.f, 0.f};
    v8f accI = {0.f, 0.f, 0.f, 0.f, 0.f, 0.f, 0.f, 0.f};
#pragma unroll
    for (int k0 = 0; k0 < 68; k0 += 4) {
      const int n0 = k0 + ko, n1 = n0 + 1;
      v2f aR, aI, bR, bI, bIn;
      aR[0] = (n0 < NMP) ? inRe[abase + (size_t)n0 * sA] : 0.f;
      aR[1] = (n1 < NMP) ? inRe[abase + (size_t)n1 * sA] : 0.f;
      aI[0] = (n0 < NMP) ? inIm[abase + (size_t)n0 * sA] : 0.f;
      aI[1] = (n1 < NMP) ? inIm[abase + (size_t)n1 * sA] : 0.f;
      bR[0] = sDre[drow + n0];
      bR[1] = sDre[drow + n1];
      bI[0] = sDim[drow + n0];
      bI[1] = sDim[drow + n1];
      bIn[0] = -bI[0];
      bIn[1] = -bI[1];
      accR = __builtin_amdgcn_wmma_f32_16x16x4_f32(false, aR, false, bR, (short)0, accR, false, false);
      accR = __builtin_amdgcn_wmma_f32_16x16x4_f32(false, aI, false, bIn, (short)0, accR, false, false);
      accI = __builtin_amdgcn_wmma_f32_16x16x4_f32(false, aR, false, bI, (short)0, accI, false, false);
      accI = __builtin_amdgcn_wmma_f32_16x16x4_f32(false, aI, false, bR, (short)0, accI, false, false);
    }
    const int cOut = ctile * 16 + lm;
    const int rhi = ((threadIdx.x >> 4) & 1) << 3;
    const int rBase = mtile * 16 + rhi;
    int u0 = rBase / NM;
    int v0 = rBase - u0 * NM;
#pragma unroll
    for (int q = 0; q < 8; ++q) {
      int vq = v0 + q;
      int uq = u0 + (vq >= NM ? 1 : 0);
      vq -= (vq >= NM ? NM : 0);
      if ((rBase + q) < NM2 && cOut < NM) {
        size_t o = vb + (size_t)uq * sU + (size_t)vq * sV + (size_t)cOut * sA;
        outRe[o] = accR[q];
        outIm[o] = accI[q];
      }
    }
  }
}

// =====================================================================
// Spectral multiply: out[2b+c] = m_c * F[b] * deconv^2
// =====================================================================
__global__ void spectral_mult(const float* __restrict__ inRe, const float* __restrict__ inIm,
                              float* __restrict__ outRe, float* __restrict__ outIm,
                              const float* __restrict__ m0r, const float* __restrict__ m0i,
                              const float* __restrict__ m1r, const float* __restrict__ m1i) {
  size_t t = (size_t)blockIdx.x * blockDim.x + threadIdx.x;
  const size_t total = (size_t)BATCH * NM * NM2;
  if (t >= total) return;
  int b = (int)(t / (NM * NM2));
  int idx = (int)(t - (size_t)b * (NM * NM2));
  int i = idx / NM2;
  int rem = idx - i * NM2;
  int j = rem / NM, k = rem - j * NM;

  float ki = (float)(i - 32), kj = (float)(j - 32), kk = (float)(k - 32);
  float k2 = ki * ki + kj * kj + kk * kk;
  float cdec = powf(PI_F / TAU_F, 1.5f);
  float d = cdec * expf(k2 * TAU_F);
  float d2 = d * d;

  size_t off = (size_t)b * VOLP + (size_t)i * (NMP * NMP) + (size_t)j * NMP + k;
  float fr = inRe[off], fi = inIm[off];

  float mr = m0r[idx], mi = m0i[idx];
  size_t o0 = (size_t)(2 * b) * VOLP + (size_t)i * (NMP * NMP) + (size_t)j * NMP + k;
  outRe[o0] = (mr * fr - mi * fi) * d2;
  outIm[o0] = (mr * fi + mi * fr) * d2;

  mr = m1r[idx];
  mi = m1i[idx];
  size_t o1 = o0 + VOLP;
  outRe[o1] = (mr * fr - mi * fi) * d2;
  outIm[o1] = (mr * fi + mi * fr) * d2;
}

// =====================================================================
// Gather stage 1 (WMMA f32): s[z=(b,c)][p][k] = sum_ij gxy[b][p][ij]*irfft[z][ij][k]
// Incremental (u,v)->offset tracking: no div/mod in the K loop.
// =====================================================================
__global__ void gather_wmma(const float* __restrict__ GXY, const float* __restrict__ Re,
                            float* __restrict__ S) {
  const int mtile = blockIdx.x, ctile = blockIdx.y, z = blockIdx.z;
  const int b = z >> 1;
  const int lm = threadIdx.x & 15;
  const int ko = (threadIdx.x >> 4) << 1;
  const float* A = GXY + (size_t)b * PPTS * IJP;   // [p][ij]
  const size_t vb = (size_t)z * VOLP;
  const int am = mtile * 16 + lm;                  // p row
  const int bcol = ctile * 16 + lm;                // k col (<80)

  // incremental offset for kk = k0 + ko: off = u*66*66 + v*66 + bcol
  int vcur = ko;
  int off0 = ko * NMP + bcol;

  v8f acc = {0.f, 0.f, 0.f, 0.f, 0.f, 0.f, 0.f, 0.f};
#pragma unroll 4
  for (int k0 = 0; k0 < IJP; k0 += 4) {
    const int n0 = k0 + ko, n1 = n0 + 1;
    v2f a, bb;
    a[0] = A[(size_t)am * IJP + n0];
    a[1] = A[(size_t)am * IJP + n1];
    int off1 = off0 + (vcur == NM - 1 ? (NMP * NMP) - (NM - 1) * NMP : NMP);
    bb[0] = Re[vb + off0];
    bb[1] = Re[vb + off1];
    acc = __builtin_amdgcn_wmma_f32_16x16x4_f32(false, a, false, bb, (short)0, acc,
                                                false, false);
    int vnext = vcur + 4;
    int wrap = (vnext >= NM) ? 1 : 0;
    off0 += 4 * NMP + wrap * ((NMP * NMP) - NM * NMP);
    vcur = vnext - wrap * NM;
  }
  const int cOut = bcol;
  const int rhi = (threadIdx.x >> 4) << 3;
#pragma unroll
  for (int q = 0; q < 8; ++q) {
    int rOut = mtile * 16 + q + rhi;   // p < 256 always
    if (cOut < NM) {
      S[((size_t)z * PPTS + rOut) * 80 + cOut] = acc[q];
    }
  }
}

// =====================================================================
// Gather stage 2: fmm[b][p][c] = (1/65^3) * sum_k s[b,c,p,k]*gz[b,p,k]
// =====================================================================
__global__ void gather_final(const float* __restrict__ S, const float* __restrict__ G,
                             float* __restrict__ out) {
  int t = blockIdx.x * blockDim.x + threadIdx.x;
  if (t >= BATCH * PPTS * 2) return;
  int b = t / (PPTS * 2);
  int rem = t - b * (PPTS * 2);
  int p = rem >> 1, c = rem & 1;
  const float* gz = G + ((size_t)(b * 3 + 2) * PPTS + p) * 80;
  const float* s = S + ((size_t)(b * 2 + c) * PPTS + p) * 80;
  float acc = 0.f;
  for (int k = 0; k < NM; ++k) acc += s[k] * gz[k];
  out[t] = acc * (1.0f / 274625.0f);   // scale = (2*pi*N/L)^3 = 65^3
}

// =====================================================================
// Launch
// =====================================================================
extern "C" void kernel_launch(void* const* d_in, const int* in_sizes, int n_in,
                              void* d_out, int out_size, void* d_ws, size_t ws_size,
                              hipStream_t stream) {
  (void)in_sizes; (void)n_in; (void)out_size; (void)ws_size;
  const float* pts = (const float*)d_in[0];
  const float* m0r = (const float*)d_in[1];
  const float* m0i = (const float*)d_in[2];
  const float* m1r = (const float*)d_in[3];
  const float* m1i = (const float*)d_in[4];
  float* ws = (float*)d_ws;
  float* out = (float*)d_out;

  float* DFre = ws + OFF_DFRE;
  float* DFim = ws + OFF_DFIM;
  float* DIre = ws + OFF_DIRE;
  float* DIim = ws + OFF_DIIM;
  float* G    = ws + OFF_G;
  float* GXY  = ws + OFF_GXY;
  float* Are  = ws + OFF_BARE;
  float* Aim  = ws + OFF_BAIM;
  float* Bre  = ws + OFF_BBRE;
  float* Bim  = ws + OFF_BBIM;
  float* S    = ws + OFF_S;

  const int S0 = NMP * NMP, S1 = NMP, S2 = 1;
  const int TB = (NTILES + 7) / 8;   // 166 blocks of 8 wave-tiles

  init_twiddles<<<(TWLD * TWLD + 255) / 256, 256, 0, stream>>>(ws);
  zero_kernel<<<4096, 256, 0, stream>>>(Are, (size_t)4 * 16 * VOLP);

  compute_g<<<(BATCH * PPTS + 127) / 128, 128, 0, stream>>>(pts, G);
  compute_gxy<<<BATCH * PPTS, 256, 0, stream>>>(G, GXY);

  // spread -> bufA Re (vols 0..7), Im stays zero
  spread_wmma<<<dim3(TB, 1, BATCH), 256, 0, stream>>>(GXY, G, Are);

  // forward shifted DFT: axis k, j, i -> result in bufB
  dft_pass_wmma<<<dim3(TB, 1, BATCH), 256, 0, stream>>>(Are, Aim, Bre, Bim, DFre, DFim, S2, S0, S1);
  dft_pass_wmma<<<dim3(TB, 1, BATCH), 256, 0, stream>>>(Bre, Bim, Are, Aim, DFre, DFim, S1, S0, S2);
  dft_pass_wmma<<<dim3(TB, 1, BATCH), 256, 0, stream>>>(Are, Aim, Bre, Bim, DFre, DFim, S0, S1, S2);

  // spectral multiply (x deconv^2, x m0/m1): bufB (8 vols) -> bufA (16 vols)
  {
    size_t total = (size_t)BATCH * NM * NM2;
    spectral_mult<<<(unsigned)((total + 255) / 256), 256, 0, stream>>>(
        Bre, Bim, Are, Aim, m0r, m0i, m1r, m1i);
  }

  // inverse shifted IDFT over 16 volumes: axis k, j, i -> bufB (Re = irfft)
  dft_pass_wmma<<<dim3(TB, 1, 16), 256, 0, stream>>>(Are, Aim, Bre, Bim, DIre, DIim, S2, S0, S1);
  dft_pass_wmma<<<dim3(TB, 1, 16), 256, 0, stream>>>(Bre, Bim, Are, Aim, DIre, DIim, S1, S0, S2);
  dft_pass_wmma<<<dim3(TB, 1, 16), 256, 0, stream>>>(Are, Aim, Bre, Bim, DIre, DIim, S0, S1, S2);

  // gather
  gather_wmma<<<dim3(16, 5, 16), 32, 0, stream>>>(GXY, Bre, S);
  gather_final<<<(BATCH * PPTS * 2 + 255) / 256, 256, 0, stream>>>(S, G, out);
}